// CrystalGNN_35562329211281
// MI455X (gfx1250) — compile-verified
//
#include <hip/hip_runtime.h>
#include <math.h>

// CGCNN forward for MI455X (gfx1250, wave32).
// Edge-GEMMs algebraically folded into node-level GEMMs (f32 WMMA 16x16x4).
// Fused 4-way GEMM writes interleaved P[n] = [Af | As | Bf | Bs] so the
// bandwidth-bound edge pass gathers two contiguous 512B segments per edge.

typedef __attribute__((ext_vector_type(2))) float v2f;
typedef __attribute__((ext_vector_type(8))) float v8f;

#define FDIM 64

__device__ __forceinline__ float sigmoid_(float x) {
    return 1.0f / (1.0f + expf(-x));
}
__device__ __forceinline__ float softplus_(float x) {
    // stable: log(1+exp(x)) = max(x,0) + log1p(exp(-|x|))
    return fmaxf(x, 0.0f) + log1pf(expf(-fabsf(x)));
}

// ---------------------------------------------------------------- zero fill
__global__ void k_zero(float* __restrict__ p, long n) {
    long i = (long)blockIdx.x * blockDim.x + threadIdx.x;
    long stride = (long)gridDim.x * blockDim.x;
    for (; i < n; i += stride) p[i] = 0.0f;
}

// ------------------------------------------------- input GEMM: [N,12]@[12,64]
__global__ void k_in_gemm(const float* __restrict__ x, const float* __restrict__ W,
                          const float* __restrict__ b, float* __restrict__ out, int nrows) {
    int idx = blockIdx.x * blockDim.x + threadIdx.x;
    if (idx >= nrows * FDIM) return;
    int n = idx >> 6, c = idx & 63;
    const float* xr = x + (long)n * 12;
    float acc = b[c];
#pragma unroll
    for (int k = 0; k < 12; ++k) acc += xr[k] * W[k * FDIM + c];
    out[idx] = acc;
}

// ------------------------------------------ BN: per-channel sum & sum-of-squares
__global__ void k_bn_reduce(const float* __restrict__ a, float* __restrict__ sums, int total) {
    __shared__ float ss[FDIM];
    __shared__ float sq[FDIM];
    if (threadIdx.x < FDIM) { ss[threadIdx.x] = 0.0f; sq[threadIdx.x] = 0.0f; }
    __syncthreads();
    int stride = gridDim.x * blockDim.x;
    for (int i = blockIdx.x * blockDim.x + threadIdx.x; i < total; i += stride) {
        float v = a[i];
        atomicAdd(&ss[i & 63], v);
        atomicAdd(&sq[i & 63], v * v);
    }
    __syncthreads();
    if (threadIdx.x < FDIM) {
        atomicAdd(&sums[threadIdx.x], ss[threadIdx.x]);
        atomicAdd(&sums[FDIM + threadIdx.x], sq[threadIdx.x]);
    }
}

__global__ void k_bn_finalize(const float* __restrict__ sums, const float* __restrict__ g,
                              const float* __restrict__ b, float* __restrict__ scsh, float inv_n) {
    int c = threadIdx.x;
    if (c >= FDIM) return;
    float mu  = sums[c] * inv_n;
    float var = sums[FDIM + c] * inv_n - mu * mu;
    float rs  = rsqrtf(var + 1e-5f);
    float sc  = rs * g[c];
    scsh[c]        = sc;
    scsh[FDIM + c] = b[c] - mu * sc;
}

__global__ void k_bn_apply(const float* __restrict__ a, const float* __restrict__ resid,
                           const float* __restrict__ scsh, float* __restrict__ out,
                           int total, int do_relu) {
    int i = blockIdx.x * blockDim.x + threadIdx.x;
    if (i >= total) return;
    int c = i & 63;
    float v = a[i] * scsh[c] + scsh[FDIM + c];
    if (resid) v += resid[i];
    if (do_relu) v = fmaxf(v, 0.0f);
    out[i] = v;
}

// -------- fused node GEMM: P[N,256] = H[N,64] @ [Wf_lo | Ws_lo | Wf_hi | Ws_hi]
// One wave -> 16 rows x 16 cols x 4 weight matrices; A fragment loaded once,
// 4 WMMAs per K-step (64 v_wmma per wave, 256 per block).
__global__ void __launch_bounds__(128) k_node_gemm4(const float* __restrict__ H,
                                                    const float* __restrict__ Wf,
                                                    const float* __restrict__ Ws,
                                                    float* __restrict__ P, int nrows) {
    int lane = threadIdx.x & 31;
    int wave = threadIdx.x >> 5;        // 0..3 -> column tile
    int row0 = blockIdx.x * 16;
    int col0 = wave * 16;
    int m  = lane & 15;
    int hi = lane >> 4;                 // 0: K pair {0,1}; 1: K pair {2,3}

    int r = row0 + m;
    if (r >= nrows) r = nrows - 1;      // clamp loads; EXEC stays all-ones
    const float* hrow = H + (long)r * FDIM + 2 * hi;

    // B column pointers: [Af]=Wf rows 0-63, [As]=Ws rows 0-63,
    //                    [Bf]=Wf rows 64-127, [Bs]=Ws rows 64-127
    const float* c0 = Wf + col0 + m + 2 * hi * FDIM;
    const float* c1 = Ws + col0 + m + 2 * hi * FDIM;
    const float* c2 = c0 + 64 * FDIM;
    const float* c3 = c1 + 64 * FDIM;

    v8f a0 = {}, a1 = {}, a2 = {}, a3 = {};
#pragma unroll
    for (int k0 = 0; k0 < FDIM; k0 += 4) {
        v2f a;  a.x  = hrow[k0];        a.y  = hrow[k0 + 1];
        v2f b0; b0.x = c0[k0 * FDIM];   b0.y = c0[k0 * FDIM + FDIM];
        v2f b1; b1.x = c1[k0 * FDIM];   b1.y = c1[k0 * FDIM + FDIM];
        v2f b2; b2.x = c2[k0 * FDIM];   b2.y = c2[k0 * FDIM + FDIM];
        v2f b3; b3.x = c3[k0 * FDIM];   b3.y = c3[k0 * FDIM + FDIM];
        a0 = __builtin_amdgcn_wmma_f32_16x16x4_f32(false, a, false, b0, (short)0, a0, false, false);
        a1 = __builtin_amdgcn_wmma_f32_16x16x4_f32(false, a, false, b1, (short)0, a1, false, false);
        a2 = __builtin_amdgcn_wmma_f32_16x16x4_f32(false, a, false, b2, (short)0, a2, false, false);
        a3 = __builtin_amdgcn_wmma_f32_16x16x4_f32(false, a, false, b3, (short)0, a3, false, false);
    }
    // C/D layout: VGPR v -> M = v + 8*hi, N = col0 + (lane&15)
#pragma unroll
    for (int v = 0; v < 8; ++v) {
        int rr = row0 + v + 8 * hi;
        if (rr < nrows) {
            long base = (long)rr * 256 + col0 + m;
            P[base]       = a0[v];   // Af
            P[base + 64]  = a1[v];   // As
            P[base + 128] = a2[v];   // Bf
            P[base + 192] = a3[v];   // Bs
        }
    }
}

// ---------------- edge pass: gate + scatter-add (one wave per edge, 2 ch/lane)
// Gathers two contiguous 512B rows: P[dst][0:128] and P[src][128:256].
__global__ void __launch_bounds__(256) k_edges(
    const int* __restrict__ src, const int* __restrict__ dst, const float* __restrict__ ea,
    const float* __restrict__ P,
    const float* __restrict__ wfE, const float* __restrict__ bf,
    const float* __restrict__ wsE, const float* __restrict__ bs,
    float* __restrict__ agg, int nedges) {
    int e = blockIdx.x * 8 + (threadIdx.x >> 5);
    if (e >= nedges) return;
    int lane = threadIdx.x & 31;
    int c = lane * 2;
    int s = src[e], d = dst[e];
    float w = ea[e];

    const float* pd = P + (long)d * 256;        // [Af | As]
    const float* ps = P + (long)s * 256 + 128;  // [Bf | Bs]

    float2 af  = *(const float2*)(pd + c);
    float2 as  = *(const float2*)(pd + 64 + c);
    float2 bfv = *(const float2*)(ps + c);
    float2 bsv = *(const float2*)(ps + 64 + c);
    float2 wf2 = *(const float2*)(wfE + c);
    float2 bf2 = *(const float2*)(bf + c);
    float2 ws2 = *(const float2*)(wsE + c);
    float2 bs2 = *(const float2*)(bs + c);

    float f0 = af.x + bfv.x + w * wf2.x + bf2.x;
    float f1 = af.y + bfv.y + w * wf2.y + bf2.y;
    float t0 = as.x + bsv.x + w * ws2.x + bs2.x;
    float t1 = as.y + bsv.y + w * ws2.y + bs2.y;

    float m0 = sigmoid_(f0) * softplus_(t0);
    float m1 = sigmoid_(f1) * softplus_(t1);

    long doff = (long)d * FDIM + c;
    atomicAdd(&agg[doff],     m0);
    atomicAdd(&agg[doff + 1], m1);
}

// ----------------------------------------------------------- mean pool (atomic)
__global__ void k_pool(const float* __restrict__ h, const int* __restrict__ batch,
                       float* __restrict__ pooled, float* __restrict__ counts, int nrows) {
    int i = blockIdx.x * blockDim.x + threadIdx.x;
    if (i >= nrows * FDIM) return;
    int n = i >> 6, c = i & 63;
    int b = batch[n];
    atomicAdd(&pooled[(long)b * FDIM + c], h[i]);
    if (c == 0) atomicAdd(&counts[b], 1.0f);
}

// -------------------------------------------------------------- MLP head [G,1]
__global__ void __launch_bounds__(32) k_head(const float* __restrict__ pooled,
                                             const float* __restrict__ counts,
                                             const float* __restrict__ W1, const float* __restrict__ b1,
                                             const float* __restrict__ W2, const float* __restrict__ b2,
                                             float* __restrict__ out, int ngraphs) {
    int g = blockIdx.x;
    if (g >= ngraphs) return;
    __shared__ float sp[FDIM];
    __shared__ float sr[32];
    int t = threadIdx.x;
    float cnt = fmaxf(counts[g], 1.0f);
    sp[t]      = pooled[(long)g * FDIM + t]      / cnt;
    sp[t + 32] = pooled[(long)g * FDIM + t + 32] / cnt;
    __syncthreads();
    float r = b1[t];
    for (int c = 0; c < FDIM; ++c) r += sp[c] * W1[c * 32 + t];
    r = fmaxf(r, 0.0f);
    sr[t] = r * W2[t];
    __syncthreads();
    if (t == 0) {
        float acc = b2[0];
        for (int j = 0; j < 32; ++j) acc += sr[j];
        out[g] = acc;
    }
}

extern "C" void kernel_launch(void* const* d_in, const int* in_sizes, int n_in,
                              void* d_out, int out_size, void* d_ws, size_t ws_size,
                              hipStream_t stream) {
    (void)n_in; (void)ws_size;
    const float* x     = (const float*)d_in[0];
    const int*   ei    = (const int*)  d_in[1];
    const float* ea    = (const float*)d_in[2];
    const int*   batch = (const int*)  d_in[3];
    const float* W_in  = (const float*)d_in[4];
    const float* b_in  = (const float*)d_in[5];
    const float* g0    = (const float*)d_in[6];
    const float* beta0 = (const float*)d_in[7];
    const float *Wf[3], *bfv[3], *Ws[3], *bsv[3], *gl[3], *bbl[3];
    for (int l = 0; l < 3; ++l) {
        int base = 8 + 6 * l;
        Wf[l]  = (const float*)d_in[base + 0];
        bfv[l] = (const float*)d_in[base + 1];
        Ws[l]  = (const float*)d_in[base + 2];
        bsv[l] = (const float*)d_in[base + 3];
        gl[l]  = (const float*)d_in[base + 4];
        bbl[l] = (const float*)d_in[base + 5];
    }
    const float* W1 = (const float*)d_in[26];
    const float* b1 = (const float*)d_in[27];
    const float* W2 = (const float*)d_in[28];
    const float* b2 = (const float*)d_in[29];

    const int N = in_sizes[0] / 12;
    const int E = in_sizes[1] / 2;
    const int G = out_size;
    const int* src = ei;
    const int* dst = ei + E;

    const long N64 = (long)N * FDIM;
    float* ws = (float*)d_ws;
    float* bufA   = ws;                 // h (ping)
    float* bufB   = ws + N64;           // agg / h_next (pong)
    float* P      = ws + 2 * N64;       // interleaved [Af|As|Bf|Bs], 4*N64
    float* stats  = ws + 6 * N64;       // 128 floats (sum, sumsq)
    float* scsh   = stats + 128;        // 128 floats (scale, shift)
    float* pooled = stats + 256;        // G*64
    float* counts = pooled + (long)G * FDIM;  // G

    const int total = N * FDIM;
    const int tB = 256;
    const int gEl  = (total + tB - 1) / tB;
    const int gRed = 1024;
    const int gGemm = (N + 15) / 16;
    const int gEdge = (E + 7) / 8;

    // ---- input layer: GEMM -> BN -> ReLU
    k_in_gemm<<<gEl, tB, 0, stream>>>(x, W_in, b_in, bufB, N);
    k_zero<<<1, 128, 0, stream>>>(stats, 128);
    k_bn_reduce<<<gRed, tB, 0, stream>>>(bufB, stats, total);
    k_bn_finalize<<<1, 64, 0, stream>>>(stats, g0, beta0, scsh, 1.0f / (float)N);
    k_bn_apply<<<gEl, tB, 0, stream>>>(bufB, nullptr, scsh, bufA, total, 1);

    float* h = bufA;
    float* o = bufB;
    for (int l = 0; l < 3; ++l) {
        // fused node-level GEMMs (WMMA f32 16x16x4) -> interleaved P
        k_node_gemm4<<<gGemm, 128, 0, stream>>>(h, Wf[l], Ws[l], P, N);
        // scatter-add of gated messages
        k_zero<<<2048, tB, 0, stream>>>(o, N64);
        k_edges<<<gEdge, tB, 0, stream>>>(src, dst, ea, P,
                                          Wf[l] + 128 * 64, bfv[l],
                                          Ws[l] + 128 * 64, bsv[l], o, E);
        // BN + residual (+ReLU for layers 0,1)
        k_zero<<<1, 128, 0, stream>>>(stats, 128);
        k_bn_reduce<<<gRed, tB, 0, stream>>>(o, stats, total);
        k_bn_finalize<<<1, 64, 0, stream>>>(stats, gl[l], bbl[l], scsh, 1.0f / (float)N);
        k_bn_apply<<<gEl, tB, 0, stream>>>(o, h, scsh, o, total, (l < 2) ? 1 : 0);
        float* t = h; h = o; o = t;
    }

    // ---- mean pool + head
    k_zero<<<64, tB, 0, stream>>>(pooled, (long)G * FDIM + G);
    k_pool<<<gEl, tB, 0, stream>>>(h, batch, pooled, counts, N);
    k_head<<<G, 32, 0, stream>>>(pooled, counts, W1, b1, W2, b2, (float*)d_out, G);
}